// CCMLite_807453851999
// MI455X (gfx1250) — compile-verified
//
#include <hip/hip_runtime.h>

// CCM-Lite fused: GroupNorm + low-rank channel mix, fp32 WMMA (16x16x4) path.
// B=16, C=256, H=W=64 (HW=4096), groups=32 (8 ch/grp), rank=12.

#define B_    16
#define C_    256
#define HW_   4096
#define G_    32
#define CPG_  8
#define R_    12
#define PROW_ (C_ * R_ * 2 + C_)   // 6400 floats per batch row of ccm_params
#define VOFF_ (C_ * R_)            // 3072
#define SOFF_ (2 * C_ * R_)        // 6144
#define EPS_  1e-6f

typedef float v2f __attribute__((ext_vector_type(2)));
typedef float v8f __attribute__((ext_vector_type(8)));

// ---------------------------------------------------------------------------
// Kernel 1: per-(b,g) mean / rstd over 8*4096 = 32768 elements.
// One block per (b,g); channels of a group are contiguous, so the block's
// region is one contiguous 128 KB span -> float4 coalesced loads.
// ---------------------------------------------------------------------------
__global__ __launch_bounds__(256) void ccm_stats(const float* __restrict__ x,
                                                 float* __restrict__ mean_ws,
                                                 float* __restrict__ rstd_ws) {
  __shared__ float s_sum[256];
  __shared__ float s_sq[256];
  const int bg = blockIdx.x;          // b*32 + g
  const int t  = threadIdx.x;
  const float4* p = (const float4*)(x + (size_t)bg * (CPG_ * HW_));
  float sum = 0.f, sq = 0.f;
#pragma unroll
  for (int i = 0; i < 32; ++i) {
    float4 v = p[i * 256 + t];
    sum += v.x + v.y + v.z + v.w;
    sq  += v.x * v.x + v.y * v.y + v.z * v.z + v.w * v.w;
  }
  s_sum[t] = sum;
  s_sq[t]  = sq;
  __syncthreads();
  for (int s = 128; s > 0; s >>= 1) {
    if (t < s) {
      s_sum[t] += s_sum[t + s];
      s_sq[t]  += s_sq[t + s];
    }
    __syncthreads();
  }
  if (t == 0) {
    const float invN = 1.0f / (float)(CPG_ * HW_);
    float mean = s_sum[0] * invN;
    float var  = s_sq[0] * invN - mean * mean;   // biased var (matches jnp.var)
    mean_ws[bg] = mean;
    rstd_ws[bg] = rsqrtf(var + EPS_);
  }
}

// ---------------------------------------------------------------------------
// Kernel 2: vtx[b,r,n] = sum_c v'[c,r]*x[b,c,n] - K[b,r]
//   v'[c,r] = v[b,c,r]*rstd[g(c)],  K[r] = sum_c v'[c,r]*mean[g(c)]
// One wave per 16-column n-tile; K=256 contraction = 64 WMMA f32 16x16x4
// steps split into two independent accumulator chains.
// WMMA layouts (32-bit, wave32): lane = h*16 + m (h=lane>>4):
//   A vgpr j  <-> A[m][2h+j]     (A = v'^T, 16r x 4c)
//   B vgpr j  <-> B[2h+j][n]     (B = x,    4c  x 16n)
//   D vgpr i  <-> D[i + 8h][n]
// ---------------------------------------------------------------------------
__global__ __launch_bounds__(256) void ccm_vtx(const float* __restrict__ x,
                                               const float* __restrict__ ccm,
                                               const float* __restrict__ mean_ws,
                                               const float* __restrict__ rstd_ws,
                                               float* __restrict__ vtx) {
  __shared__ float s_v[C_ * 16];   // v' padded to 16 rows (rows 12..15 = 0)
  __shared__ float s_mean[C_];
  __shared__ float s_K[16];

  const int b     = blockIdx.x >> 5;   // 32 chunks of 128 columns per batch
  const int chunk = blockIdx.x & 31;
  const int t     = threadIdx.x;

  {  // build v' (scaled by rstd) in LDS; one thread per channel
    const int   c  = t;
    const float rs = rstd_ws[b * G_ + (c >> 3)];
    s_mean[c]      = mean_ws[b * G_ + (c >> 3)];
    const float* vp = ccm + (size_t)b * PROW_ + VOFF_ + c * R_;
#pragma unroll
    for (int r = 0; r < R_; ++r) s_v[c * 16 + r] = vp[r] * rs;
#pragma unroll
    for (int r = R_; r < 16; ++r) s_v[c * 16 + r] = 0.f;
  }
  __syncthreads();
  if (t < 16) {  // K[r] correction term (tiny serial reduction)
    float k = 0.f;
    if (t < R_) {
      for (int c = 0; c < C_; ++c) k += s_v[c * 16 + t] * s_mean[c];
    }
    s_K[t] = k;
  }
  __syncthreads();

  const int wave = t >> 5;
  const int lane = t & 31;
  const int h    = lane >> 4;
  const int m    = lane & 15;                 // r-row for A, n-col for B/D
  const int n0   = (chunk * 8 + wave) * 16;
  const float* xb = x + (size_t)b * C_ * HW_ + n0 + m;

  v8f acc0 = {0.f, 0.f, 0.f, 0.f, 0.f, 0.f, 0.f, 0.f};
  v8f acc1 = {0.f, 0.f, 0.f, 0.f, 0.f, 0.f, 0.f, 0.f};
#pragma unroll 4
  for (int i = 0; i < 32; ++i) {
    const int cA = i * 4 + 2 * h;          // chain 0: channels 0..127
    const int cB = 128 + i * 4 + 2 * h;    // chain 1: channels 128..255
    v2f a0, a1, b0, b1;
    a0.x = s_v[(cA + 0) * 16 + m];
    a0.y = s_v[(cA + 1) * 16 + m];
    a1.x = s_v[(cB + 0) * 16 + m];
    a1.y = s_v[(cB + 1) * 16 + m];
    b0.x = xb[(size_t)(cA + 0) * HW_];
    b0.y = xb[(size_t)(cA + 1) * HW_];
    b1.x = xb[(size_t)(cB + 0) * HW_];
    b1.y = xb[(size_t)(cB + 1) * HW_];
    acc0 = __builtin_amdgcn_wmma_f32_16x16x4_f32(false, a0, false, b0,
                                                 (short)0, acc0, false, false);
    acc1 = __builtin_amdgcn_wmma_f32_16x16x4_f32(false, a1, false, b1,
                                                 (short)0, acc1, false, false);
  }
  v8f acc = acc0 + acc1;

  float* vout = vtx + (size_t)b * R_ * HW_ + n0 + m;
#pragma unroll
  for (int i = 0; i < 8; ++i) {
    const int M = i + 8 * h;     // r index of this D row
    if (M < R_) vout[(size_t)M * HW_] = acc[i] - s_K[M];
  }
}

// ---------------------------------------------------------------------------
// Kernel 3: out[b,c,n] = (x - mean)*rstd + sum_r u[c,r]*vtx[r,n] + shift[c]
// One wave per 16c x 16n output tile: K=12 = exactly 3 WMMA 16x16x4 steps,
// then fused GroupNorm + shift epilogue (x read once, out written once).
// D row i+8h spans channels c0+8h .. c0+8h+7 -> one 8-channel group per lane.
// ---------------------------------------------------------------------------
__global__ __launch_bounds__(256) void ccm_out(const float* __restrict__ x,
                                               const float* __restrict__ ccm,
                                               const float* __restrict__ mean_ws,
                                               const float* __restrict__ rstd_ws,
                                               const float* __restrict__ vtx,
                                               float* __restrict__ out) {
  const int t    = threadIdx.x;
  const int wave = t >> 5;
  const int lane = t & 31;
  const int h    = lane >> 4;
  const int m    = lane & 15;

  const int gw  = blockIdx.x * 8 + wave;   // 0 .. 65535
  const int b   = gw >> 12;                // 4096 tiles per batch
  const int rem = gw & 4095;
  const int c0  = (rem >> 8) << 4;         // 16 c-tiles
  const int n0  = (rem & 255) << 4;        // 256 n-tiles

  const float* ub  = ccm + (size_t)b * PROW_;               // u[c][r], r fastest
  const float* vtb = vtx + (size_t)b * R_ * HW_ + n0 + m;

  v8f acc = {0.f, 0.f, 0.f, 0.f, 0.f, 0.f, 0.f, 0.f};
#pragma unroll
  for (int kb = 0; kb < R_; kb += 4) {
    const int k = kb + 2 * h;
    v2f a, bb;
    a.x  = ub[(c0 + m) * R_ + k + 0];
    a.y  = ub[(c0 + m) * R_ + k + 1];
    bb.x = vtb[(size_t)(k + 0) * HW_];
    bb.y = vtb[(size_t)(k + 1) * HW_];
    acc = __builtin_amdgcn_wmma_f32_16x16x4_f32(false, a, false, bb,
                                                (short)0, acc, false, false);
  }

  const int   crow = c0 + 8 * h;                 // first channel of this half
  const float mean = mean_ws[b * G_ + (crow >> 3)];
  const float rstd = rstd_ws[b * G_ + (crow >> 3)];
  const float* xb  = x   + (size_t)(b * C_ + crow) * HW_ + n0 + m;
  float*       ob  = out + (size_t)(b * C_ + crow) * HW_ + n0 + m;
  const float* sh  = ccm + (size_t)b * PROW_ + SOFF_ + crow;
#pragma unroll
  for (int i = 0; i < 8; ++i) {
    float xv = xb[(size_t)i * HW_];
    ob[(size_t)i * HW_] = (xv - mean) * rstd + acc[i] + sh[i];
  }
}

// ---------------------------------------------------------------------------
extern "C" void kernel_launch(void* const* d_in, const int* in_sizes, int n_in,
                              void* d_out, int out_size, void* d_ws, size_t ws_size,
                              hipStream_t stream) {
  (void)in_sizes; (void)n_in; (void)out_size; (void)ws_size;
  const float* x   = (const float*)d_in[0];
  const float* ccm = (const float*)d_in[1];
  float*       out = (float*)d_out;

  // workspace: [mean 512][rstd 512][vtx 16*12*4096]  (~3.1 MB)
  float* mean_ws = (float*)d_ws;
  float* rstd_ws = mean_ws + B_ * G_;
  float* vtx     = rstd_ws + B_ * G_;

  ccm_stats<<<B_ * G_, 256, 0, stream>>>(x, mean_ws, rstd_ws);
  ccm_vtx  <<<B_ * G_, 256, 0, stream>>>(x, ccm, mean_ws, rstd_ws, vtx);
  ccm_out  <<<8192,    256, 0, stream>>>(x, ccm, mean_ws, rstd_ws, vtx, out);
}